// InterpretedFlockingModel_53644141527383
// MI455X (gfx1250) — compile-verified
//
#include <hip/hip_runtime.h>

// ---------------------------------------------------------------------------
// CDNA5 async global->LDS path (probe-confirmed on this toolchain).
// Lowers to GLOBAL_LOAD_ASYNC_TO_LDS_B128 (ASYNCcnt) + S_WAIT_ASYNCCNT.
// ---------------------------------------------------------------------------
#if defined(__gfx1250__) && \
    __has_builtin(__builtin_amdgcn_global_load_async_to_lds_b128) && \
    __has_builtin(__builtin_amdgcn_s_wait_asynccnt)
#define HAVE_ASYNC_LDS 1
#else
#define HAVE_ASYNC_LDS 0
#endif

#if HAVE_ASYNC_LDS
typedef int v4i __attribute__((vector_size(16)));
typedef __attribute__((address_space(1))) v4i* g_v4i_p;   // global int4*
typedef __attribute__((address_space(3))) v4i* l_v4i_p;   // LDS int4*
#define ASYNC_B128(gp, lp) \
  __builtin_amdgcn_global_load_async_to_lds_b128((g_v4i_p)(gp), (l_v4i_p)(lp), 0, 0)
#define WAIT_ASYNC(n) __builtin_amdgcn_s_wait_asynccnt(n)
#endif

#define CHUNK 256            // edges per chunk per wave
#define WAVES_PER_BLOCK 8    // 256 threads

// Native fp32 atomic add (global_atomic_add_f32), not the CAS-loop fallback.
__device__ __forceinline__ void fatomic_add(float* p, float v) {
#if defined(__AMDGCN__)
  unsafeAtomicAdd(p, v);
#else
  atomicAdd(p, v);
#endif
}

// ---------------------------------------------------------------------------
// probe_kernel: NEVER LAUNCHED. Defined first so the disasm snippet shows,
// in one screen: global_load_async_to_lds_b128, s_wait_asynccnt, a ds_load,
// and the lowering of unsafeAtomicAdd (want: global_atomic_add_f32).
// ---------------------------------------------------------------------------
__global__ void probe_kernel(const long long* __restrict__ g,
                             float* __restrict__ out) {
#if HAVE_ASYNC_LDS
  __shared__ long long buf[64];
  int lane = threadIdx.x & 31;
  ASYNC_B128((const char*)g + lane * 16, (char*)buf + lane * 16);
  WAIT_ASYNC(0);
  asm volatile("" ::: "memory");
  int v = ((const int*)buf)[2 * lane];
  fatomic_add(&out[v & 63], 1.0f);
#else
  fatomic_add(&out[threadIdx.x & 63], 1.0f);
#endif
}

// ---------------------------------------------------------------------------
// Per-edge work.  acc is AoS: float4 per node = [sum(m0), sum(m1), sum(m3), cnt]
// so all 4 atomics of one edge hit the same 16B region / L2 line.
// (sum(m2) = aggr[0] = y4 is dead in the node update -> eliminated.)
// ---------------------------------------------------------------------------
__device__ __forceinline__ void process_edge(int s, int d,
                                             const float2* __restrict__ pos2,
                                             const float2* __restrict__ vel2,
                                             float* __restrict__ acc) {
  float2 pd = pos2[d];
  float2 ps = pos2[s];
  float2 vd = vel2[d];
  float2 vs = vel2[s];

  float* node = acc + 4 * (unsigned)d;
  fatomic_add(node + 3, 1.0f);  // cnt counts every edge, even "same" ones

  float ddx = ps.x - pd.x;            // x4 - x0
  float ddy = pd.y - ps.y;            // x1 - x5
  bool same = (ddx == 0.0f) && (ddy == 0.0f) &&
              (vs.x - vd.x == 0.0f) && (vs.y - vd.y == 0.0f);
  if (!same) {
    float m0 = (ddx + ddy * 0.40914905f) * 0.028998906f;
    float m1 = (ddx + ddy * 0.5819344f) * -0.02637788f;
    float m3 = (pd.y * 0.95594215f - ps.y - pd.x * 0.20244296f +
                ps.x * 0.17809269f) * 0.026933579f;
    fatomic_add(node + 0, m0);
    fatomic_add(node + 1, m1);
    fatomic_add(node + 2, m3);
  }
}

#if HAVE_ASYNC_LDS
// Stage one 256-edge chunk of src + dst indices (int64) into this wave's LDS
// buffer: 2048 B per index array = 4 x (32 lanes x 16 B) async b128 issues each.
__device__ __forceinline__ void issue_chunk(const long long* __restrict__ gsrc,
                                            const long long* __restrict__ gdst,
                                            long long* lbuf, int lane) {
  const char* gs = (const char*)gsrc;
  const char* gd = (const char*)gdst;
  char* lsrc = (char*)lbuf;
  char* ldst = (char*)(lbuf + CHUNK);
#pragma unroll
  for (int r = 0; r < 4; ++r) {
    int off = lane * 16 + r * 512;
    ASYNC_B128(gs + off, lsrc + off);
    ASYNC_B128(gd + off, ldst + off);
  }
}

__device__ __forceinline__ void process_lds_chunk(const long long* lbuf, int lane,
                                                  const float2* __restrict__ pos2,
                                                  const float2* __restrict__ vel2,
                                                  float* __restrict__ acc) {
  // indices are < 100000 -> low dword of each int64 (little endian)
  const int* ls = (const int*)lbuf;
  const int* ld = (const int*)(lbuf + CHUNK);
#pragma unroll
  for (int i = 0; i < CHUNK / 32; ++i) {
    int k = lane + i * 32;                 // stride-2-dword LDS reads: conflict-free
    int s = ls[2 * k];
    int d = ld[2 * k];
    process_edge(s, d, pos2, vel2, acc);
  }
}
#endif  // HAVE_ASYNC_LDS

// ---------------------------------------------------------------------------
// Kernel: edge scatter. Per-wave double-buffered async staging of the 102 MB
// edge stream; node gathers hit L2 (node table = 1.6 MB).
// ---------------------------------------------------------------------------
__global__ __launch_bounds__(256) void edge_kernel(const float* __restrict__ pos,
                                                   const float* __restrict__ vel,
                                                   const long long* __restrict__ ei,
                                                   int E, int n,
                                                   float* __restrict__ acc) {
  const float2* pos2 = (const float2*)pos;
  const float2* vel2 = (const float2*)vel;
  const long long* src = ei;
  const long long* dst = ei + E;

  const int nChunks = E / CHUNK;

#if HAVE_ASYNC_LDS
  const int lane = threadIdx.x & 31;
  const int wave = threadIdx.x >> 5;
  const int gwave = blockIdx.x * WAVES_PER_BLOCK + wave;
  const int nWaves = gridDim.x * WAVES_PER_BLOCK;

  // 8 waves x 2 buffers x (256 src + 256 dst) int64 = 64 KB LDS
  __shared__ long long smem[WAVES_PER_BLOCK * 2 * (2 * CHUNK)];
  long long* buf0 = &smem[(wave * 2 + 0) * (2 * CHUNK)];
  long long* buf1 = &smem[(wave * 2 + 1) * (2 * CHUNK)];

  int c = gwave;
  int cur = 0;
  if (c < nChunks) issue_chunk(src + (size_t)c * CHUNK, dst + (size_t)c * CHUNK, buf0, lane);
  for (; c < nChunks; c += nWaves) {
    long long* pb = cur ? buf1 : buf0;
    long long* nb = cur ? buf0 : buf1;
    int nxt = c + nWaves;
    if (nxt < nChunks) {
      issue_chunk(src + (size_t)nxt * CHUNK, dst + (size_t)nxt * CHUNK, nb, lane);
      WAIT_ASYNC(8);   // drain current chunk's 8 issues; next chunk stays in flight
    } else {
      WAIT_ASYNC(0);
    }
    asm volatile("" ::: "memory");  // keep LDS reads below the async wait
    process_lds_chunk(pb, lane, pos2, vel2, acc);
    cur ^= 1;
  }
#else
  int tid = blockIdx.x * blockDim.x + threadIdx.x;
  int nt = gridDim.x * blockDim.x;
  for (int e = tid; e < nChunks * CHUNK; e += nt) {
    process_edge((int)src[e], (int)dst[e], pos2, vel2, acc);
  }
#endif

  // tail (E % CHUNK) via direct loads (empty for E = 6.4M)
  int tid2 = blockIdx.x * blockDim.x + threadIdx.x;
  int nt2 = gridDim.x * blockDim.x;
  for (int e = nChunks * CHUNK + tid2; e < E; e += nt2) {
    process_edge((int)src[e], (int)dst[e], pos2, vel2, acc);
  }
}

// ---------------------------------------------------------------------------
// Kernel: zero accumulators (d_ws is poisoned by harness; must re-zero
// every launch for determinism).
// ---------------------------------------------------------------------------
__global__ __launch_bounds__(256) void zero_kernel(float* __restrict__ p, int n) {
  int i = blockIdx.x * blockDim.x + threadIdx.x;
  if (i < n) p[i] = 0.0f;
}

// ---------------------------------------------------------------------------
// Kernel: per-node pointwise update.  acc[i] = [sum(m0), sum(m1), sum(m3), cnt]
// ---------------------------------------------------------------------------
__global__ __launch_bounds__(256) void node_kernel(const float* __restrict__ pos,
                                                   const float* __restrict__ vel,
                                                   const float4* __restrict__ acc,
                                                   float* __restrict__ out, int n) {
  int i = blockIdx.x * blockDim.x + threadIdx.x;
  if (i >= n) return;
  float y0 = pos[2 * i];
  float y1 = pos[2 * i + 1];
  float y2 = vel[2 * i];
  float y3 = vel[2 * i + 1];

  float4 a = acc[i];
  float cnt = a.w;
  float inv = cnt > 0.0f ? 1.0f / cnt : 0.0f;   // cnt is an integer count
  float y5 = a.z;                                // add_out1  = sum(m3)
  float y6 = a.x * inv;                          // mean_out0 = mean(m0)
  float y7 = a.y * inv;                          // mean_out1 = mean(m1)

  float u0 = (y1 + y7 / 0.037233025f) * -0.0020586958f;
  float u1 = ((y0 - y7 * y7 * y5) * 0.015168043f - y6) * -0.10450508f;
  float u2 = (y7 - y1 * 0.027931638f + y6) * 0.075265266f;
  float u3 = (y3 * y3 - y6 + y2 - y3 + y7 * -0.33928046f) * -0.08554904f;

  float p0 = (u0 + u3 + u0) * -0.24326763f - u1 / 0.7301285f - u2 * 1.1234615f;
  float p1 = u2 - (u1 + u0) + u3;

  out[2 * i]     = p0;
  out[2 * i + 1] = p1;
}

// ---------------------------------------------------------------------------
// Launch
// ---------------------------------------------------------------------------
extern "C" void kernel_launch(void* const* d_in, const int* in_sizes, int n_in,
                              void* d_out, int out_size, void* d_ws, size_t ws_size,
                              hipStream_t stream) {
  const float* pos = (const float*)d_in[0];
  const float* vel = (const float*)d_in[1];
  const long long* ei = (const long long*)d_in[2];   // int64 edge_index (2, E)

  int n = in_sizes[0] / 2;   // 100000 nodes
  int E = in_sizes[2] / 2;   // 6.4M edges
  float* acc = (float*)d_ws; // n x float4: [sum(m0), sum(m1), sum(m3), cnt]
  float* out = (float*)d_out;

  int accN = 4 * n;
  zero_kernel<<<(accN + 255) / 256, 256, 0, stream>>>(acc, accN);

  edge_kernel<<<1024, 256, 0, stream>>>(pos, vel, ei, E, n, acc);

  node_kernel<<<(n + 255) / 256, 256, 0, stream>>>(pos, vel, (const float4*)acc, out, n);
}